// FusionBlock_DenseAVInteractions_27444841022213
// MI455X (gfx1250) — compile-verified
//
#include <hip/hip_runtime.h>
#include <math.h>

typedef __attribute__((ext_vector_type(2))) float v2f;
typedef __attribute__((ext_vector_type(8))) float v8f;
typedef __attribute__((ext_vector_type(4))) unsigned int u32x4;
typedef __attribute__((ext_vector_type(8))) unsigned int u32x8;

#define B_    2
#define N_MM  256
#define N_V   64
#define N_A   128
#define DIM   1024
#define HEADS 16
#define INNER 256
#define HID   4096
#define HD    16
#define SCALE 0.125f   /* (DIM/HEADS)^-0.5 = 64^-0.5 */
#define EPS   1e-5f

#define EPI_NONE      0
#define EPI_BIAS_RES  2
#define EPI_BIAS_GELU 3

// Tensor Data Mover path for LDS staging (fallback: set to 0 -> manual loads)
#define USE_TDM 1

// ------------------------------------------------------------------
// LayerNorm over rows of length 1024 (one block per row, 256 threads)
// ------------------------------------------------------------------
__global__ __launch_bounds__(256) void ln1024_kernel(const float* __restrict__ x,
                                                     const float* __restrict__ w,
                                                     const float* __restrict__ b,
                                                     float* __restrict__ y) {
  __shared__ float red[256];
  const int row = blockIdx.x;
  const float* xr = x + (long)row * DIM;
  float v[4];
  float s = 0.f;
#pragma unroll
  for (int i = 0; i < 4; ++i) { v[i] = xr[threadIdx.x + 256 * i]; s += v[i]; }
  red[threadIdx.x] = s;
  __syncthreads();
  for (int off = 128; off > 0; off >>= 1) {
    if (threadIdx.x < off) red[threadIdx.x] += red[threadIdx.x + off];
    __syncthreads();
  }
  const float mean = red[0] * (1.f / DIM);
  __syncthreads();
  float s2 = 0.f;
#pragma unroll
  for (int i = 0; i < 4; ++i) { float d = v[i] - mean; s2 += d * d; }
  red[threadIdx.x] = s2;
  __syncthreads();
  for (int off = 128; off > 0; off >>= 1) {
    if (threadIdx.x < off) red[threadIdx.x] += red[threadIdx.x + off];
    __syncthreads();
  }
  const float inv = rsqrtf(red[0] * (1.f / DIM) + EPS);
  float* yr = y + (long)row * DIM;
#pragma unroll
  for (int i = 0; i < 4; ++i) {
    const int c = threadIdx.x + 256 * i;
    yr[c] = (v[i] - mean) * inv * w[c] + b[c];
  }
}

#if USE_TDM
// ------------------------------------------------------------------
// Issue one TDM 2-D tile load: tensor row-major [tensor_rows x row_len],
// tile [tile_rows x tile_cols] starting at gptr, packed row-major at lptr.
// D# layout per CDNA5 ISA ch.8: group0 = count/lds_addr/global_addr/type=2,
// group1 = data_size, tensor dims, tile dims, dim0 stride.
// ------------------------------------------------------------------
__device__ __forceinline__ void tdm_load_2d(const float* gptr, const float* lptr,
                                            unsigned tile_cols, unsigned tile_rows,
                                            unsigned row_len, unsigned tensor_rows) {
  const unsigned long long ga = (unsigned long long)gptr;
  // generic LDS pointer: aperture in high 32 bits, LDS byte offset in addr[31:0]
  const unsigned la = (unsigned)(unsigned long long)lptr;
  u32x4 g0;
  g0[0] = 1u;                                               // count=1, user mode
  g0[1] = la;                                               // lds_addr (bytes)
  g0[2] = (unsigned)(ga & 0xFFFFFFFFull);                   // global_addr[31:0]
  g0[3] = (unsigned)((ga >> 32) & 0x1FFFFFFull) | (2u << 30); // addr[56:32] | type=2
  u32x8 g1;
  g1[0] = 2u << 16;                                         // data_size=2 (4 bytes)
  g1[1] = (row_len & 0xFFFFu) << 16;                        // tensor_dim0[15:0]
  g1[2] = ((row_len >> 16) & 0xFFFFu) | ((tensor_rows & 0xFFFFu) << 16); // dim0 hi | dim1 lo
  g1[3] = ((tensor_rows >> 16) & 0xFFFFu) | (tile_cols << 16);           // dim1 hi | tile_dim0
  g1[4] = tile_rows;                                        // tile_dim1 (tile_dim2=0)
  g1[5] = row_len;                                          // tensor_dim0_stride[31:0]
  g1[6] = 0u;                                               // stride0 hi | stride1 lo
  g1[7] = 0u;                                               // stride1 hi
  asm volatile("tensor_load_to_lds %0, %1" :: "s"(g0), "s"(g1) : "memory");
}
#endif

// ------------------------------------------------------------------
// fp32 WMMA GEMM:  C[M,N] = A[M,K] @ B[K,N]  (+ bias / residual / gelu)
// Block = 8 wave32 (256 thr), tile 32(M) x 128(N), K panels of 64 in LDS.
// TDM (wave 0) DMA-fills double-buffered LDS panels; s_wait_tensorcnt 2
// overlaps the next panel's DMA with WMMA on the current one.
// M%32==0, N%128==0, K%64==0 at every launch site -> no divergent
// branches around WMMA (EXEC all-ones, as the ISA requires).
// ------------------------------------------------------------------
__global__ __launch_bounds__(256) void gemm_wmma_f32(
    const float* __restrict__ A, const float* __restrict__ Bm,
    const float* __restrict__ bias, const float* __restrict__ res,
    float* __restrict__ C, int M, int N, int K, int epi) {
  __shared__ float lA[2][32 * 64];     // [m][k] panels
  __shared__ float lB[2][64 * 128];    // [k][n] panels
  const int wave = threadIdx.x >> 5;
  const int lane = threadIdx.x & 31;
  const int m0   = blockIdx.x * 32;
  const int nblk = blockIdx.y * 128;
  const int n0   = nblk + wave * 16;

  v8f acc0 = {0.f, 0.f, 0.f, 0.f, 0.f, 0.f, 0.f, 0.f};
  v8f acc1 = acc0;

  const int S = K >> 6;  // number of 64-deep K panels

#if USE_TDM
  if (wave == 0) {       // TDM ignores EXEC; issue once per block
    tdm_load_2d(&A[(long)m0 * K], lA[0], 64u, 32u, (unsigned)K, (unsigned)M);
    tdm_load_2d(&Bm[nblk],        lB[0], 128u, 64u, (unsigned)N, (unsigned)K);
  }
#endif

  for (int i = 0; i < S; ++i) {
    const int cur = i & 1;
#if USE_TDM
    if (wave == 0) {
      if (i + 1 < S) {   // prefetch next panel into the other buffer
        const int kn = (i + 1) << 6;
        tdm_load_2d(&A[(long)m0 * K + kn],  lA[cur ^ 1], 64u, 32u, (unsigned)K, (unsigned)M);
        tdm_load_2d(&Bm[(long)kn * N + nblk], lB[cur ^ 1], 128u, 64u, (unsigned)N, (unsigned)K);
        asm volatile("s_wait_tensorcnt 0x2" ::: "memory");  // panel i done, i+1 in flight
      } else {
        asm volatile("s_wait_tensorcnt 0x0" ::: "memory");
      }
    }
    __syncthreads();     // publish panel i to all 8 waves
#else
    const int k0 = i << 6;
    __syncthreads();
    for (int t = threadIdx.x; t < 32 * 64; t += 256) {
      const int r = t >> 6, c = t & 63;
      lA[cur][t] = A[(long)(m0 + r) * K + (k0 + c)];
    }
    for (int t = threadIdx.x; t < 64 * 128; t += 256) {
      const int r = t >> 7, c = t & 127;
      lB[cur][t] = Bm[(long)(k0 + r) * N + (nblk + c)];
    }
    __syncthreads();
#endif

    const float* lAc = lA[cur];
    const float* lBc = lB[cur];
    const int ml    = lane & 15;          // A row in tile / B col in tile
    const int khalf = (lane >> 4) << 1;   // lanes 16-31 carry K+2,K+3
#pragma unroll
    for (int kk = 0; kk < 64; kk += 4) {
      const int kb = kk + khalf;
      v2f bf;  // B 4x16 frag: VGPR0 = B[kb][n], VGPR1 = B[kb+1][n]
      bf.x = lBc[(kb + 0) * 128 + wave * 16 + ml];
      bf.y = lBc[(kb + 1) * 128 + wave * 16 + ml];
      v2f a0, a1;  // A 16x4 frags, M rows 0-15 and 16-31 of block tile
      a0.x = lAc[ml * 64 + kb];            a0.y = lAc[ml * 64 + kb + 1];
      a1.x = lAc[(ml + 16) * 64 + kb];     a1.y = lAc[(ml + 16) * 64 + kb + 1];
      acc0 = __builtin_amdgcn_wmma_f32_16x16x4_f32(false, a0, false, bf,
                                                   (short)0, acc0, false, false);
      acc1 = __builtin_amdgcn_wmma_f32_16x16x4_f32(false, a1, false, bf,
                                                   (short)0, acc1, false, false);
    }
    __syncthreads();     // all waves done with panel i before its buffer refills
  }

  // C/D layout: VGPR r -> row r (lanes 0-15) or r+8 (lanes 16-31), col = lane&15
  const int col   = n0 + (lane & 15);
  const int rbase = (lane >> 4) << 3;
  const float bv  = (epi != EPI_NONE) ? bias[col] : 0.f;
#pragma unroll
  for (int r = 0; r < 8; ++r) {
    const int row0 = m0 + rbase + r;
    const int row1 = row0 + 16;
    float x0 = acc0[r] + bv;
    float x1 = acc1[r] + bv;
    if (epi == EPI_BIAS_GELU) {
      x0 = 0.5f * x0 * (1.f + erff(x0 * 0.7071067811865476f));
      x1 = 0.5f * x1 * (1.f + erff(x1 * 0.7071067811865476f));
    } else if (epi == EPI_BIAS_RES) {
      x0 += res[(long)row0 * N + col];
      x1 += res[(long)row1 * N + col];
    }
    C[(long)row0 * N + col] = x0;
    C[(long)row1 * N + col] = x1;
  }
}

// ------------------------------------------------------------------
// Factorized attention. Logits over the synthetic 8192-long KV separate as
// s(v,a) = s_v + s_a, so softmax/PV factor into two independent softmaxes:
//   out = softmax(q.k_v^T) @ v_v  +  softmax(q.k_a^T) @ v_a
// One block per (batch, head); thread t owns q-token t. K/V tiles in LDS.
// ------------------------------------------------------------------
__global__ __launch_bounds__(256) void attn_factored_kernel(
    const float* __restrict__ q, const float* __restrict__ kvv,
    const float* __restrict__ kva, float* __restrict__ out) {
  const int b = blockIdx.x >> 4;
  const int h = blockIdx.x & 15;
  __shared__ float kV[N_V][HD], vV[N_V][HD], kA[N_A][HD], vA[N_A][HD];
  for (int i = threadIdx.x; i < N_V * HD; i += 256) {
    const int v = i >> 4, d = i & 15;
    const float* p = kvv + (long)(b * N_V + v) * (2 * INNER);
    kV[v][d] = p[h * HD + d];
    vV[v][d] = p[INNER + h * HD + d];
  }
  for (int i = threadIdx.x; i < N_A * HD; i += 256) {
    const int a = i >> 4, d = i & 15;
    const float* p = kva + (long)(b * N_A + a) * (2 * INNER);
    kA[a][d] = p[h * HD + d];
    vA[a][d] = p[INNER + h * HD + d];
  }
  __syncthreads();

  const int tok = threadIdx.x;
  float qr[HD];
  const float* qp = q + (long)(b * N_MM + tok) * INNER + h * HD;
#pragma unroll
  for (int d = 0; d < HD; ++d) qr[d] = qp[d];

  float o[HD];
#pragma unroll
  for (int d = 0; d < HD; ++d) o[d] = 0.f;

  // ---- video stream (64 keys) ----
  float mV = -3.4e38f;
  for (int v = 0; v < N_V; ++v) {
    float s = 0.f;
#pragma unroll
    for (int d = 0; d < HD; ++d) s += qr[d] * kV[v][d];
    mV = fmaxf(mV, s * SCALE);
  }
  float lV = 0.f;
  for (int v = 0; v < N_V; ++v) {
    float s = 0.f;
#pragma unroll
    for (int d = 0; d < HD; ++d) s += qr[d] * kV[v][d];
    const float e = __expf(s * SCALE - mV);
    lV += e;
#pragma unroll
    for (int d = 0; d < HD; ++d) o[d] += e * vV[v][d];
  }
  const float invV = 1.f / lV;
#pragma unroll
  for (int d = 0; d < HD; ++d) o[d] *= invV;

  // ---- audio stream (128 keys) ----
  float mA = -3.4e38f;
  for (int a = 0; a < N_A; ++a) {
    float s = 0.f;
#pragma unroll
    for (int d = 0; d < HD; ++d) s += qr[d] * kA[a][d];
    mA = fmaxf(mA, s * SCALE);
  }
  float lAcc = 0.f;
  float oa[HD];
#pragma unroll
  for (int d = 0; d < HD; ++d) oa[d] = 0.f;
  for (int a = 0; a < N_A; ++a) {
    float s = 0.f;
#pragma unroll
    for (int d = 0; d < HD; ++d) s += qr[d] * kA[a][d];
    const float e = __expf(s * SCALE - mA);
    lAcc += e;
#pragma unroll
    for (int d = 0; d < HD; ++d) oa[d] += e * vA[a][d];
  }
  const float invA = 1.f / lAcc;

  float* op = out + (long)(b * N_MM + tok) * INNER + h * HD;
#pragma unroll
  for (int d = 0; d < HD; ++d) op[d] = o[d] + oa[d] * invA;
}

// ------------------------------------------------------------------
extern "C" void kernel_launch(void* const* d_in, const int* in_sizes, int n_in,
                              void* d_out, int out_size, void* d_ws, size_t ws_size,
                              hipStream_t stream) {
  (void)in_sizes; (void)n_in; (void)out_size; (void)ws_size;
  const float* xmm      = (const float*)d_in[0];
  const float* xv       = (const float*)d_in[1];
  const float* xa       = (const float*)d_in[2];
  const float* ln_mm_w  = (const float*)d_in[3];
  const float* ln_mm_b  = (const float*)d_in[4];
  const float* ln_v_w   = (const float*)d_in[5];
  const float* ln_v_b   = (const float*)d_in[6];
  const float* ln_a_w   = (const float*)d_in[7];
  const float* ln_a_b   = (const float*)d_in[8];
  const float* Wq       = (const float*)d_in[9];
  const float* Wkv      = (const float*)d_in[10];
  const float* Wproj    = (const float*)d_in[11];
  const float* bproj    = (const float*)d_in[12];
  const float* ln_mlp_w = (const float*)d_in[13];
  const float* ln_mlp_b = (const float*)d_in[14];
  const float* W1       = (const float*)d_in[15];
  const float* b1       = (const float*)d_in[16];
  const float* W2       = (const float*)d_in[17];
  const float* b2       = (const float*)d_in[18];
  float* out = (float*)d_out;
  float* ws  = (float*)d_ws;

  const int RM = B_ * N_MM;  // 512 rows (mm stream)
  float* src_mm = ws;                               // 512*1024
  float* src_v  = src_mm + (long)RM * DIM;          // 128*1024
  float* src_a  = src_v + (long)B_ * N_V * DIM;     // 256*1024
  float* qbuf   = src_a + (long)B_ * N_A * DIM;     // 512*256
  float* kvv    = qbuf + (long)RM * INNER;          // 128*512
  float* kva    = kvv + (long)B_ * N_V * 2 * INNER; // 256*512
  float* attnb  = kva + (long)B_ * N_A * 2 * INNER; // 512*256
  float* xmm2   = attnb + (long)RM * INNER;         // 512*1024
  float* hln    = xmm2 + (long)RM * DIM;            // 512*1024
  float* act    = hln + (long)RM * DIM;             // 512*4096

  // 1) LayerNorms
  ln1024_kernel<<<RM, 256, 0, stream>>>(xmm, ln_mm_w, ln_mm_b, src_mm);
  ln1024_kernel<<<B_ * N_V, 256, 0, stream>>>(xv, ln_v_w, ln_v_b, src_v);
  ln1024_kernel<<<B_ * N_A, 256, 0, stream>>>(xa, ln_a_w, ln_a_b, src_a);

  // 2) Projections (WMMA GEMMs). Wkv[:DIM] / Wkv[DIM:] are contiguous halves.
  gemm_wmma_f32<<<dim3(16, 2), 256, 0, stream>>>(src_mm, Wq, nullptr, nullptr,
                                                 qbuf, RM, INNER, DIM, EPI_NONE);
  gemm_wmma_f32<<<dim3(4, 4), 256, 0, stream>>>(src_v, Wkv, nullptr, nullptr,
                                                kvv, B_ * N_V, 2 * INNER, DIM, EPI_NONE);
  gemm_wmma_f32<<<dim3(8, 4), 256, 0, stream>>>(src_a, Wkv + (long)DIM * 2 * INNER,
                                                nullptr, nullptr,
                                                kva, B_ * N_A, 2 * INNER, DIM, EPI_NONE);

  // 3) Factorized attention over the separable 8192-key softmax
  attn_factored_kernel<<<B_ * HEADS, 256, 0, stream>>>(qbuf, kvv, kva, attnb);

  // 4) Output proj + bias + residual -> xmm2
  gemm_wmma_f32<<<dim3(16, 8), 256, 0, stream>>>(attnb, Wproj, bproj, xmm,
                                                 xmm2, RM, DIM, INNER, EPI_BIAS_RES);

  // 5) MLP: LN -> W1+gelu -> W2 + residual
  ln1024_kernel<<<RM, 256, 0, stream>>>(xmm2, ln_mlp_w, ln_mlp_b, hln);
  gemm_wmma_f32<<<dim3(16, 32), 256, 0, stream>>>(hln, W1, b1, nullptr,
                                                  act, RM, HID, DIM, EPI_BIAS_GELU);
  gemm_wmma_f32<<<dim3(16, 8), 256, 0, stream>>>(act, W2, b2, xmm2,
                                                 out, RM, DIM, HID, EPI_BIAS_RES);
}